// Siddon_69088843923456
// MI455X (gfx1250) — compile-verified
//
#include <hip/hip_runtime.h>
#include <math.h>

// Siddon ray projector for MI455X (gfx1250, wave32).
//  - volume + mask (128 MB) are L2-resident (192 MB L2) -> gather-latency bound.
//  - 3-way merge march (no sort), restricted to the ray/box alpha window.
//  - 8 lanes cooperate per ray (chunked alpha window) -> 8192 waves in flight.
//  - global_prefetch_b8 lookahead along the ray; shfl_xor (ds path) reduction.

#define XD 256            // cubic volume dimension
#define NPLANE XD         // plane indices 0..XD (257 planes / axis)
#define KSPLIT 8          // lanes per ray
#define BIGA   3.0e38f
#define BIGH   1.0e38f

__global__ __launch_bounds__(256)
void siddon_march_kernel(const float* __restrict__ vol,
                         const float* __restrict__ src,
                         const float* __restrict__ tgt,
                         const float* __restrict__ msk,
                         float* __restrict__ out,
                         int BD, int D)
{
    const int gid = blockIdx.x * blockDim.x + threadIdx.x;
    const int ray = gid >> 3;           // / KSPLIT
    const int j   = gid & (KSPLIT - 1); // chunk id within ray
    if (ray >= BD) return;
    const int b = ray / D;
    const int d = ray - b * D;

    // ---- ray setup (redundant across the 8 lanes of a ray; cheap) ----
    const int base = ray * 3;
    const float sx = src[base + 0], sy = src[base + 1], sz = src[base + 2];
    const float dx = tgt[base + 0] - sx + 1e-8f;
    const float dy = tgt[base + 1] - sy + 1e-8f;
    const float dz = tgt[base + 2] - sz + 1e-8f;
    const float raylen = sqrtf(fmaf(dx, dx, fmaf(dy, dy, dz * dz)));

    const float ivx = 1.0f / dx, ivy = 1.0f / dy, ivz = 1.0f / dz;
    // alpha of plane i on axis a:  (i - 0.5 - s_a) * iv_a  = fmaf(i, iv, cb)
    const float cbx = (-0.5f - sx) * ivx;
    const float cby = (-0.5f - sy) * ivy;
    const float cbz = (-0.5f - sz) * ivz;
    const float lastx = fmaf((float)NPLANE, ivx, cbx);  // alpha at plane XD
    const float lasty = fmaf((float)NPLANE, ivy, cby);
    const float lastz = fmaf((float)NPLANE, ivz, cbz);

    // ray/box alpha window
    const float ain = fmaxf(fmaxf(fminf(cbx, lastx), fminf(cby, lasty)),
                            fminf(cbz, lastz));
    const float aout = fminf(fminf(fmaxf(cbx, lastx), fmaxf(cby, lasty)),
                             fmaxf(cbz, lastz));

    float acc0 = 0.f, acc1 = 0.f, acc2 = 0.f, acc3 = 0.f;

    if (aout > ain) {
        // pad the window by ~2 crossings so boundary midpoints are kept;
        // validity test still zeroes anything truly outside.
        const float delta = 2.0f * fmaxf(fmaxf(fabsf(ivx), fabsf(ivy)), fabsf(ivz));
        const float start = ain - delta;
        const float cw = (aout + delta - start) * (1.0f / (float)KSPLIT);
        const float c0 = fmaf((float)j, cw, start);
        const float c1 = c0 + cw;

        // first crossing with alpha >= c0, per axis (O(1) chunk entry)
        const int stx = (dx > 0.f) ? 1 : -1;
        const int sty = (dy > 0.f) ? 1 : -1;
        const int stz = (dz > 0.f) ? 1 : -1;
        const float p0x = fmaf(c0, dx, sx);
        const float p0y = fmaf(c0, dy, sy);
        const float p0z = fmaf(c0, dz, sz);
        int ixx = (dx > 0.f) ? (int)ceilf(p0x + 0.5f) : (int)floorf(p0x + 0.5f);
        int iyy = (dy > 0.f) ? (int)ceilf(p0y + 0.5f) : (int)floorf(p0y + 0.5f);
        int izz = (dz > 0.f) ? (int)ceilf(p0z + 0.5f) : (int)floorf(p0z + 0.5f);

        float ax = (ixx >= 0 && ixx <= NPLANE) ? fmaf((float)ixx, ivx, cbx) : BIGA;
        float ay = (iyy >= 0 && iyy <= NPLANE) ? fmaf((float)iyy, ivy, cby) : BIGA;
        float az = (izz >= 0 && izz <= NPLANE) ? fmaf((float)izz, ivz, cbz) : BIGA;

        #define ADVANCE_MIN()                                                          \
            if (ax <= ay && ax <= az) {                                                \
                ixx += stx;                                                            \
                ax = (ixx >= 0 && ixx <= NPLANE) ? fmaf((float)ixx, ivx, cbx) : BIGA;  \
            } else if (ay <= az) {                                                     \
                iyy += sty;                                                            \
                ay = (iyy >= 0 && iyy <= NPLANE) ? fmaf((float)iyy, ivy, cby) : BIGA;  \
            } else {                                                                   \
                izz += stz;                                                            \
                az = (izz >= 0 && izz <= NPLANE) ? fmaf((float)izz, ivz, cbz) : BIGA;  \
            }

        float prev = fminf(fminf(ax, ay), az);
        if (prev < BIGH) {
            ADVANCE_MIN();
            const float SCL = (float)(XD - 1) / (float)XD;   // grid_sample scaling
            int iter = 0;
            while (prev < c1 && iter < 1024) {
                const float cur = fminf(fminf(ax, ay), az);
                if (cur >= BIGH) break;

                const float dl = cur - prev;
                const float amid = 0.5f * (prev + cur);
                const float px = fmaf(amid, dx, sx);
                const float py = fmaf(amid, dy, sy);
                const float pz = fmaf(amid, dz, sz);
                const float fx = rintf(px * SCL);   // v_rndne: half-to-even, matches jnp.round
                const float fy = rintf(py * SCL);
                const float fz = rintf(pz * SCL);
                const bool valid =
                    (fx >= 0.f) & (fx <= (float)(XD - 1)) &
                    (fy >= 0.f) & (fy <= (float)(XD - 1)) &
                    (fz >= 0.f) & (fz <= (float)(XD - 1));
                if (valid) {
                    const int lin = (((int)fx * XD) + (int)fy) * XD + (int)fz;
                    const float v = vol[lin];
                    int c = (int)msk[lin];
                    c = (c < 0) ? 0 : ((c > 3) ? 3 : c);
                    const float w = v * dl;
                    acc0 += (c == 0) ? w : 0.f;
                    acc1 += (c == 1) ? w : 0.f;
                    acc2 += (c == 2) ? w : 0.f;
                    acc3 += (c == 3) ? w : 0.f;

                    // gfx1250 prefetch (global_prefetch_b8): ~12 voxels ahead
                    if ((iter & 3) == 0) {
                        const float la = 12.0f * dl;
                        float qx = fminf(fmaxf(rintf(fmaf(la, dx, px) * SCL), 0.f), (float)(XD - 1));
                        float qy = fminf(fmaxf(rintf(fmaf(la, dy, py) * SCL), 0.f), (float)(XD - 1));
                        float qz = fminf(fmaxf(rintf(fmaf(la, dz, pz) * SCL), 0.f), (float)(XD - 1));
                        const int lin2 = (((int)qx * XD) + (int)qy) * XD + (int)qz;
                        __builtin_prefetch(vol + lin2, 0, 3);
                        __builtin_prefetch(msk + lin2, 0, 3);
                    }
                }
                ADVANCE_MIN();
                prev = cur;
                ++iter;
            }
        }
        #undef ADVANCE_MIN
    }

    // ---- reduce the 8 chunk-partials of this ray (wave32 lane permute / ds path) ----
    #pragma unroll
    for (int off = KSPLIT >> 1; off >= 1; off >>= 1) {
        acc0 += __shfl_xor(acc0, off, KSPLIT);
        acc1 += __shfl_xor(acc1, off, KSPLIT);
        acc2 += __shfl_xor(acc2, off, KSPLIT);
        acc3 += __shfl_xor(acc3, off, KSPLIT);
    }

    if (j == 0) {
        // out shape (B, 4, D); every (b,c,d) written exactly once
        float* o = out + (size_t)b * 4 * D + d;
        o[0 * D] = acc0 * raylen;
        o[1 * D] = acc1 * raylen;
        o[2 * D] = acc2 * raylen;
        o[3 * D] = acc3 * raylen;
    }
}

extern "C" void kernel_launch(void* const* d_in, const int* in_sizes, int n_in,
                              void* d_out, int out_size, void* d_ws, size_t ws_size,
                              hipStream_t stream) {
    const float* vol = (const float*)d_in[0];   // (256,256,256)
    const float* src = (const float*)d_in[1];   // (B,D,3)
    const float* tgt = (const float*)d_in[2];   // (B,D,3)
    const float* msk = (const float*)d_in[3];   // (256,256,256)
    float* out = (float*)d_out;                 // (B,4,D)

    const int BD = in_sizes[1] / 3;             // B*D = 32768
    int D = 16384;
    if (BD < D) D = BD;                         // defensive

    const int total = BD * KSPLIT;              // 262144 threads = 8192 waves
    const int threads = 256;                    // 8 wave32 per block
    const int blocks = (total + threads - 1) / threads;
    siddon_march_kernel<<<blocks, threads, 0, stream>>>(vol, src, tgt, msk, out, BD, D);
}